// Encode_all_27006754357381
// MI455X (gfx1250) — compile-verified
//
#include <hip/hip_runtime.h>
#include <math.h>

typedef __attribute__((ext_vector_type(2))) float v2f;
typedef __attribute__((ext_vector_type(8))) float v8f;

// ---------------------------------------------------------------------------
// Generic fp32 WMMA GEMM: C[M x Nn] = A[M x K] @ B[K x Nn]
// One wave (32 lanes) produces one 16x64 tile of C using V_WMMA_F32_16X16X4_F32,
// reusing the A fragment across four 16-wide N-subtiles.
// Requires: M % 16 == 0 handled by Mtiles guard, K % 4 == 0, Nn % 64 == 0.
// blockDim = (32,4): 4 independent waves -> 4 row tiles per block.
// ---------------------------------------------------------------------------
__global__ void wmma_gemm_f32(const float* __restrict__ A,
                              const float* __restrict__ B,
                              float* __restrict__ C,
                              int K, int ldn /* = Nn, ld of B and C */,
                              int Mtiles) {
  const int lane  = threadIdx.x;                 // 0..31
  const int tileM = blockIdx.x * 4 + threadIdx.y;
  if (tileM >= Mtiles) return;                   // uniform per wave
  const int row0 = tileM * 16;
  const int col0 = blockIdx.y * 64;
  const int mr    = lane & 15;                   // M (for A) / N (for B) index
  const int khalf = (lane >> 4) << 1;            // lanes 16-31 handle K+2,K+3

  v8f acc0 = {}, acc1 = {}, acc2 = {}, acc3 = {};
  const float* Arow = A + (size_t)(row0 + mr) * K + khalf;

  for (int k = 0; k < K; k += 4) {
    v2f a;
    a.x = Arow[k + 0];
    a.y = Arow[k + 1];
    const float* Bk = B + (size_t)(k + khalf) * ldn + col0 + mr;
    v2f b0, b1, b2, b3;
    b0.x = Bk[0];  b0.y = Bk[ldn + 0];
    b1.x = Bk[16]; b1.y = Bk[ldn + 16];
    b2.x = Bk[32]; b2.y = Bk[ldn + 32];
    b3.x = Bk[48]; b3.y = Bk[ldn + 48];
    acc0 = __builtin_amdgcn_wmma_f32_16x16x4_f32(false, a, false, b0, (short)0, acc0, false, false);
    acc1 = __builtin_amdgcn_wmma_f32_16x16x4_f32(false, a, false, b1, (short)0, acc1, false, false);
    acc2 = __builtin_amdgcn_wmma_f32_16x16x4_f32(false, a, false, b2, (short)0, acc2, false, false);
    acc3 = __builtin_amdgcn_wmma_f32_16x16x4_f32(false, a, false, b3, (short)0, acc3, false, false);
  }

  // C/D layout: VGPR v holds M = v (lanes 0-15) / v+8 (lanes 16-31), N = lane&15
  const int mbase = row0 + ((lane >> 4) ? 8 : 0);
  float* Cp = C + (size_t)mbase * ldn + col0 + mr;
#pragma unroll
  for (int v = 0; v < 8; ++v) {
    Cp[(size_t)v * ldn + 0]  = acc0[v];
    Cp[(size_t)v * ldn + 16] = acc1[v];
    Cp[(size_t)v * ldn + 32] = acc2[v];
    Cp[(size_t)v * ldn + 48] = acc3[v];
  }
}

// ---------------------------------------------------------------------------
// COO SpMM: out[row[e], :] += val[e] * x[col[e], :]   (out pre-zeroed)
// One float4 of the D-wide row per thread; consecutive threads cover one edge
// -> coalesced gathers of x and coalesced fp32 atomic scatters.
// D = 4 << vshift.
// ---------------------------------------------------------------------------
__global__ void spmm_kernel(const int* __restrict__ rows,
                            const int* __restrict__ cols,
                            const float* __restrict__ vals,
                            const float* __restrict__ x,
                            float* __restrict__ out,
                            int E, int D, int vshift) {
  long long gid = (long long)blockIdx.x * blockDim.x + threadIdx.x;
  long long total = (long long)E << vshift;
  if (gid >= total) return;
  int e  = (int)(gid >> vshift);
  int c4 = ((int)gid & ((1 << vshift) - 1)) << 2;
  int r = rows[e], c = cols[e];
  float v = vals[e];
  const float4 xv = *(const float4*)(x + (size_t)c * D + c4);
  float* o = out + (size_t)r * D + c4;
  atomicAdd(o + 0, v * xv.x);
  atomicAdd(o + 1, v * xv.y);
  atomicAdd(o + 2, v * xv.z);
  atomicAdd(o + 3, v * xv.w);
}

// ---------------------------------------------------------------------------
// Attention fusion. 32 lanes per node (8 nodes / 256-thread block).
// Lane owns dims {lane, lane+32}. Wave-local shuffle reduction (deterministic).
// ---------------------------------------------------------------------------
__global__ void fuse_kernel(const float* __restrict__ e_sp1, const float* __restrict__ e_sp2,
                            const float* __restrict__ e_ft1, const float* __restrict__ e_ft2,
                            const float* __restrict__ w_omega, const float* __restrict__ u_omega,
                            float* __restrict__ o_sp_comb, float* __restrict__ o_ft_comb,
                            float* __restrict__ o_comb, float* __restrict__ o_alpha, int N) {
  __shared__ float sW[64 * 64];
  __shared__ float sU[64];
  __shared__ float sE[8][2][64];
  const int tid = threadIdx.x;
  for (int i = tid; i < 64 * 64; i += 256) sW[i] = w_omega[i];
  if (tid < 64) sU[tid] = u_omega[tid];
  const int local = tid >> 5;
  const int lane  = tid & 31;
  const int node  = blockIdx.x * 8 + local;
  const bool active = node < N;                 // uniform per wave
  size_t b = (size_t)node * 64;
  if (active) {
    float sp0 = 0.5f * (e_sp1[b + lane]      + e_sp2[b + lane]);
    float sp1 = 0.5f * (e_sp1[b + lane + 32] + e_sp2[b + lane + 32]);
    float ft0 = 0.5f * (e_ft1[b + lane]      + e_ft2[b + lane]);
    float ft1 = 0.5f * (e_ft1[b + lane + 32] + e_ft2[b + lane + 32]);
    o_sp_comb[b + lane] = sp0; o_sp_comb[b + lane + 32] = sp1;
    o_ft_comb[b + lane] = ft0; o_ft_comb[b + lane + 32] = ft1;
    sE[local][0][lane] = sp0; sE[local][0][lane + 32] = sp1;
    sE[local][1][lane] = ft0; sE[local][1][lane + 32] = ft1;
  }
  __syncthreads();
  if (!active) return;
  // v[m,e] = tanh(sum_d emb[m,d] * W[d,e]); this lane handles e = lane, lane+32
  float a00 = 0.f, a01 = 0.f, a10 = 0.f, a11 = 0.f;
  for (int k = 0; k < 64; ++k) {
    float w0 = sW[k * 64 + lane];
    float w1 = sW[k * 64 + lane + 32];
    float m0 = sE[local][0][k];
    float m1 = sE[local][1][k];
    a00 += m0 * w0; a01 += m0 * w1;
    a10 += m1 * w0; a11 += m1 * w1;
  }
  float vu0 = tanhf(a00) * sU[lane] + tanhf(a01) * sU[lane + 32];
  float vu1 = tanhf(a10) * sU[lane] + tanhf(a11) * sU[lane + 32];
  for (int off = 16; off > 0; off >>= 1) {
    vu0 += __shfl_down(vu0, off, 32);
    vu1 += __shfl_down(vu1, off, 32);
  }
  vu0 = __shfl(vu0, 0, 32);
  vu1 = __shfl(vu1, 0, 32);
  float s0 = vu0 + 1e-6f, s1 = vu1 + 1e-6f;
  float mx = fmaxf(s0, s1);
  float ex0 = expf(s0 - mx), ex1 = expf(s1 - mx);
  float inv = 1.0f / (ex0 + ex1);
  float al0 = ex0 * inv, al1 = ex1 * inv;
  o_comb[b + lane]      = al0 * sE[local][0][lane]      + al1 * sE[local][1][lane];
  o_comb[b + lane + 32] = al0 * sE[local][0][lane + 32] + al1 * sE[local][1][lane + 32];
  if (lane == 0) {
    o_alpha[(size_t)node * 2 + 0] = al0;
    o_alpha[(size_t)node * 2 + 1] = al1;
  }
}

extern "C" void kernel_launch(void* const* d_in, const int* in_sizes, int n_in,
                              void* d_out, int out_size, void* d_ws, size_t ws_size,
                              hipStream_t stream) {
  const float* feat1  = (const float*)d_in[0];   // [N,1024]
  const float* feat2  = (const float*)d_in[1];   // [N,256]
  const int*   row_sp1 = (const int*)d_in[2];
  const int*   col_sp1 = (const int*)d_in[3];
  const float* val_sp1 = (const float*)d_in[4];
  const int*   row_ft1 = (const int*)d_in[5];
  const int*   col_ft1 = (const int*)d_in[6];
  const float* val_ft1 = (const float*)d_in[7];
  const int*   row_sp2 = (const int*)d_in[8];
  const int*   col_sp2 = (const int*)d_in[9];
  const float* val_sp2 = (const float*)d_in[10];
  const int*   row_ft2 = (const int*)d_in[11];
  const int*   col_ft2 = (const int*)d_in[12];
  const float* val_ft2 = (const float*)d_in[13];
  const float* w_enc1  = (const float*)d_in[14]; // [1024,64]
  const float* w_enc2  = (const float*)d_in[15]; // [256,64]
  const float* w_dec1  = (const float*)d_in[16]; // [64,1024]
  const float* w_dec2  = (const float*)d_in[17]; // [64,256]
  const float* w_omega = (const float*)d_in[18]; // [64,64]
  const float* u_omega = (const float*)d_in[19]; // [64,1]

  const int N = in_sizes[0] / 1024;
  const int E = in_sizes[2];

  float* out = (float*)d_out;
  const size_t n64 = (size_t)N * 64;
  float* o_emb_sp1 = out;
  float* o_emb_sp2 = out + n64;
  float* o_emb_ft1 = out + 2 * n64;
  float* o_emb_ft2 = out + 3 * n64;
  float* o_sp_comb = out + 4 * n64;
  float* o_ft_comb = out + 5 * n64;
  float* o_comb    = out + 6 * n64;
  float* o_rec_sp1 = out + 7 * n64;
  float* o_rec_sp2 = o_rec_sp1 + (size_t)N * 1024;
  float* o_rec_ft1 = o_rec_sp2 + (size_t)N * 256;
  float* o_rec_ft2 = o_rec_ft1 + (size_t)N * 1024;
  float* o_alpha   = o_rec_ft2 + (size_t)N * 256;

  float* ws   = (float*)d_ws;
  float* h1   = ws;                        // [N,64]
  float* h2   = h1 + n64;                  // [N,64]
  float* dec1 = h2 + n64;                  // [N,1024]
  float* dec2 = dec1 + (size_t)N * 1024;   // [N,256]

  // SpMM outputs are accumulated atomically -> zero the whole output buffer.
  hipMemsetAsync(d_out, 0, (size_t)out_size * sizeof(float), stream);

  const int Mtiles = (N + 15) / 16;        // 3125 for N=50000 (exact)
  const dim3 gblk(32, 4);
  const int gx = (Mtiles + 3) / 4;

  // Encoders: h1 = feat1 @ w_enc1, h2 = feat2 @ w_enc2
  wmma_gemm_f32<<<dim3(gx, 1), gblk, 0, stream>>>(feat1, w_enc1, h1, 1024, 64, Mtiles);
  wmma_gemm_f32<<<dim3(gx, 1), gblk, 0, stream>>>(feat2, w_enc2, h2, 256, 64, Mtiles);

  // Encoder SpMMs (D = 64, vshift = 4)
  {
    long long tot = (long long)E << 4;
    int nb = (int)((tot + 255) >> 8);
    spmm_kernel<<<nb, 256, 0, stream>>>(row_sp1, col_sp1, val_sp1, h1, o_emb_sp1, E, 64, 4);
    spmm_kernel<<<nb, 256, 0, stream>>>(row_sp2, col_sp2, val_sp2, h2, o_emb_sp2, E, 64, 4);
    spmm_kernel<<<nb, 256, 0, stream>>>(row_ft1, col_ft1, val_ft1, h1, o_emb_ft1, E, 64, 4);
    spmm_kernel<<<nb, 256, 0, stream>>>(row_ft2, col_ft2, val_ft2, h2, o_emb_ft2, E, 64, 4);
  }

  // Attention fusion
  fuse_kernel<<<(N + 7) / 8, 256, 0, stream>>>(o_emb_sp1, o_emb_sp2, o_emb_ft1, o_emb_ft2,
                                               w_omega, u_omega,
                                               o_sp_comb, o_ft_comb, o_comb, o_alpha, N);

  // Decoders: dec1 = emb_comb @ w_dec1 [N,1024]; dec2 = emb_comb @ w_dec2 [N,256]
  wmma_gemm_f32<<<dim3(gx, 16), gblk, 0, stream>>>(o_comb, w_dec1, dec1, 64, 1024, Mtiles);
  wmma_gemm_f32<<<dim3(gx, 4),  gblk, 0, stream>>>(o_comb, w_dec2, dec2, 64, 256, Mtiles);

  // Decoder SpMMs
  {
    long long tot1 = (long long)E << 8;   // D = 1024
    int nb1 = (int)((tot1 + 255) >> 8);
    long long tot2 = (long long)E << 6;   // D = 256
    int nb2 = (int)((tot2 + 255) >> 8);
    spmm_kernel<<<nb1, 256, 0, stream>>>(row_sp1, col_sp1, val_sp1, dec1, o_rec_sp1, E, 1024, 8);
    spmm_kernel<<<nb2, 256, 0, stream>>>(row_sp2, col_sp2, val_sp2, dec2, o_rec_sp2, E, 256, 6);
    spmm_kernel<<<nb1, 256, 0, stream>>>(row_ft1, col_ft1, val_ft1, dec1, o_rec_ft1, E, 1024, 8);
    spmm_kernel<<<nb2, 256, 0, stream>>>(row_ft2, col_ft2, val_ft2, dec2, o_rec_ft2, E, 256, 6);
  }
}